// DINO_7490422964299
// MI455X (gfx1250) — compile-verified
//
#include <hip/hip_runtime.h>

// ---------------- problem constants (match reference) ----------------
#define K_PROTO 65536
#define NCHUNK  256          // n = min(s_chunk, t_chunk) = 256
#define SPLIT   4            // K-dimension split per chunk
#define TPB     256          // 8 wave32 per workgroup
#define KCOLS   (K_PROTO / SPLIT)       // 16384 floats per split
#define ITERS   (KCOLS / 4 / TPB)       // 16 float4 iterations per thread

typedef float v2f __attribute__((ext_vector_type(2)));
typedef float v8f __attribute__((ext_vector_type(8)));

__device__ __forceinline__ void ld4(const float4* __restrict__ p, int i, float* o) {
    float4 t = p[i];
    o[0] = t.x; o[1] = t.y; o[2] = t.z; o[3] = t.w;
}

// online softmax accumulate, with weighted sum W += exp(x-m)*s
__device__ __forceinline__ void online_w(float x, float s, float& m, float& z, float& w) {
    if (x > m) { float r = __expf(m - x); z *= r; w *= r; m = x; }
    float e = __expf(x - m);
    z += e;
    w = __fmaf_rn(e, s, w);
}
__device__ __forceinline__ void online(float x, float& m, float& z) {
    if (x > m) { z *= __expf(m - x); m = x; }
    z += __expf(x - m);
}

// Pass 1: one workgroup per (chunk, k-split). Streams 8 rows once, online
// softmax stats, then WG reduction (max tree -> rescale -> WMMA sum).
// Writes 16 floats per WG to ws: [0]=Mt0 [1]=Mt1 [2..6]=Ms_j
//                                [7]=Zt0 [8]=Wt0 [9]=Zt1 [10]=Wt1 [11..15]=Zs_j
__global__ __launch_bounds__(TPB) void dino_pass1(
    const float* __restrict__ vs, const float* __restrict__ vt,
    const float* __restrict__ center, float* __restrict__ ws)
{
    const int chunk = blockIdx.x;   // 0..255
    const int split = blockIdx.y;   // 0..3
    const int tid   = threadIdx.x;
    const int col0  = split * KCOLS;

    const float4* cp = (const float4*)(center + col0);
    const float4* t0 = (const float4*)(vt + (size_t)(2 * chunk    ) * K_PROTO + col0);
    const float4* t1 = (const float4*)(vt + (size_t)(2 * chunk + 1) * K_PROTO + col0);
    const float4* sp[5];
#pragma unroll
    for (int j = 0; j < 5; ++j)
        sp[j] = (const float4*)(vs + (size_t)(6 * chunk + j) * K_PROTO + col0);

    float mt0 = -3.0e38f, zt0 = 0.f, wt0 = 0.f;
    float mt1 = -3.0e38f, zt1 = 0.f, wt1 = 0.f;
    float msj[5], zsj[5];
#pragma unroll
    for (int j = 0; j < 5; ++j) { msj[j] = -3.0e38f; zsj[j] = 0.f; }

    for (int it = 0; it < ITERS; ++it) {
        const int idx = it * TPB + tid;
        // software prefetch next tile (speculative; OOB dropped by HW)
        __builtin_prefetch(&cp[idx + TPB], 0, 1);
        __builtin_prefetch(&t0[idx + TPB], 0, 1);
        __builtin_prefetch(&t1[idx + TPB], 0, 1);
#pragma unroll
        for (int j = 0; j < 5; ++j) __builtin_prefetch(&sp[j][idx + TPB], 0, 1);

        float cf[4], a0f[4], a1f[4], bf[5][4];
        ld4(cp, idx, cf);
        ld4(t0, idx, a0f);
        ld4(t1, idx, a1f);
#pragma unroll
        for (int j = 0; j < 5; ++j) ld4(sp[j], idx, bf[j]);

#pragma unroll
        for (int q = 0; q < 4; ++q) {
            float xs[5];
#pragma unroll
            for (int j = 0; j < 5; ++j)
                xs[j] = __fmaf_rn(bf[j][q], 10.f, 1e-20f);   // vs/T_STUDENT + 1e-20
            const float sv = xs[0] + xs[1] + xs[2] + xs[3] + xs[4];
#pragma unroll
            for (int j = 0; j < 5; ++j) online(xs[j], msj[j], zsj[j]);
            const float u0 = (a0f[q] - cf[q]) * 25.f;        // (vt-center)/T_TEACHER
            const float u1 = (a1f[q] - cf[q]) * 25.f;
            online_w(u0, sv, mt0, zt0, wt0);
            online_w(u1, sv, mt1, zt1, wt1);
        }
    }

    // ---- stage 1: workgroup max tree over 7 channels ----
    __shared__ float smax[8 * TPB];          // [ch][tid]
    smax[0 * TPB + tid] = mt0;
    smax[1 * TPB + tid] = mt1;
#pragma unroll
    for (int j = 0; j < 5; ++j) smax[(2 + j) * TPB + tid] = msj[j];
    smax[7 * TPB + tid] = -3.0e38f;
    __syncthreads();
    for (int s = TPB / 2; s > 0; s >>= 1) {
        if (tid < s) {
#pragma unroll
            for (int ch = 0; ch < 8; ++ch)
                smax[ch * TPB + tid] = fmaxf(smax[ch * TPB + tid], smax[ch * TPB + tid + s]);
        }
        __syncthreads();
    }
    const float Mt0 = smax[0 * TPB];
    const float Mt1 = smax[1 * TPB];
    float Ms[5];
#pragma unroll
    for (int j = 0; j < 5; ++j) Ms[j] = smax[(2 + j) * TPB];

    // ---- stage 2: rescale per-thread sums to the WG max -> plain sums ----
    __shared__ float pmat[16 * TPB];         // [ch][tid]
    {
        const float r0 = __expf(mt0 - Mt0);
        const float r1 = __expf(mt1 - Mt1);
        pmat[0 * TPB + tid] = zt0 * r0;
        pmat[1 * TPB + tid] = wt0 * r0;
        pmat[2 * TPB + tid] = zt1 * r1;
        pmat[3 * TPB + tid] = wt1 * r1;
#pragma unroll
        for (int j = 0; j < 5; ++j)
            pmat[(4 + j) * TPB + tid] = zsj[j] * __expf(msj[j] - Ms[j]);
#pragma unroll
        for (int ch = 9; ch < 16; ++ch) pmat[ch * TPB + tid] = 0.f;
    }
    __syncthreads();

    // ---- stage 3: WMMA reduction. D = sum_t ones(16x4) x P(4x16) over 64 blocks.
    // A all-ones => D[m][n] = sum over all 256 t of pmat[n][t] (column sums),
    // invariant to the exact K-slot permutation inside B.
    __shared__ float sres[16];
    if (tid < 32) {                          // wave 0 only: EXEC all ones
        const int n     = tid & 15;
        const int kbase = (tid >> 4) * 2;    // VGPR0 -> K = 2*half, VGPR1 -> K = 2*half+1
        v2f a; a[0] = 1.f; a[1] = 1.f;
        v8f acc = {};
#pragma unroll
        for (int blk = 0; blk < TPB / 4; ++blk) {    // 64 chained v_wmma_f32_16x16x4_f32
            v2f b;
            b[0] = pmat[n * TPB + blk * 4 + kbase + 0];
            b[1] = pmat[n * TPB + blk * 4 + kbase + 1];
            acc = __builtin_amdgcn_wmma_f32_16x16x4_f32(
                false, a, false, b, (short)0, acc, false, false);
        }
        if (tid < 16) sres[tid] = acc[0];    // D row 0: lane n = channel-n total
    }
    __syncthreads();

    if (tid == 0) {
        float* o = ws + ((size_t)(chunk * SPLIT + split)) * 16;
        o[0] = Mt0;
        o[1] = Mt1;
#pragma unroll
        for (int j = 0; j < 5; ++j) o[2 + j] = Ms[j];
#pragma unroll
        for (int i = 0; i < 9; ++i) o[7 + i] = sres[i];
    }
}

// Pass 2: one block; thread c merges SPLIT partials of chunk c, computes the
// chunk loss, then a block-sum produces the scalar.
__global__ __launch_bounds__(TPB) void dino_pass2(
    const float* __restrict__ ws, float* __restrict__ out)
{
    const int c = threadIdx.x;               // chunk id, 0..255
    const float* p = ws + (size_t)c * (SPLIT * 16);

    float M[7];
#pragma unroll
    for (int ch = 0; ch < 7; ++ch) {
        float m = p[ch];
#pragma unroll
        for (int s = 1; s < SPLIT; ++s) m = fmaxf(m, p[s * 16 + ch]);
        M[ch] = m;
    }
    const int mc[9] = {0, 0, 1, 1, 2, 3, 4, 5, 6};   // sum-channel -> governing max
    float S[9];
#pragma unroll
    for (int i = 0; i < 9; ++i) {
        float acc = 0.f;
#pragma unroll
        for (int s = 0; s < SPLIT; ++s)
            acc = __fmaf_rn(p[s * 16 + 7 + i], __expf(p[s * 16 + mc[i]] - M[mc[i]]), acc);
        S[i] = acc;
    }
    // loss_c = sum_i W_i/Z_i - 2 * sum_j (Ms_j + log Zs_j)
    float loss_c = S[1] / S[0] + S[3] / S[2];
#pragma unroll
    for (int j = 0; j < 5; ++j) loss_c -= 2.f * (M[2 + j] + __logf(S[4 + j]));

    __shared__ float red[TPB];
    red[c] = loss_c;
    __syncthreads();
    for (int s = TPB / 2; s > 0; s >>= 1) {
        if (c < s) red[c] += red[c + s];
        __syncthreads();
    }
    if (c == 0) out[0] = -red[0] * (1.0f / 2560.0f);   // mean over 256*2*5, negated
}

extern "C" void kernel_launch(void* const* d_in, const int* in_sizes, int n_in,
                              void* d_out, int out_size, void* d_ws, size_t ws_size,
                              hipStream_t stream) {
    (void)in_sizes; (void)n_in; (void)out_size; (void)ws_size;
    const float* vs     = (const float*)d_in[0];
    const float* vt     = (const float*)d_in[1];
    const float* center = (const float*)d_in[2];
    float* ws  = (float*)d_ws;                 // needs 1024*16*4 = 64 KB
    float* out = (float*)d_out;

    dim3 grid(NCHUNK, SPLIT);
    dino_pass1<<<grid, TPB, 0, stream>>>(vs, vt, center, ws);
    dino_pass2<<<1, TPB, 0, stream>>>(ws, out);
}